// RadixAttention_23433341567745
// MI455X (gfx1250) — compile-verified
//
#include <hip/hip_runtime.h>

typedef __attribute__((ext_vector_type(2))) float v2f;
typedef __attribute__((ext_vector_type(8))) float v8f;

#define B_ 32
#define HQ_ 32
#define HKV_ 8
#define D_ 128
#define G_ 4
#define PAGE_ 16
#define PPS_ 128
#define SPLITS_ 4
#define CPS_ (PPS_ / SPLITS_) /* 32 pages (chunks) per split */
#define SCALING_ 0.08838834764831845f
#define CAP_ 50.0f

#define KROW_ 132 /* K tile row stride (floats): conflict-free b64 A-frags      */
#define VROW_ 136 /* V tile row stride (floats): 136%64==8 -> conflict-free b32 */
#define TILE_F_ (16 * KROW_ + 16 * VROW_) /* 4288 floats per tile buffer */
#define WAVE_LDS_ (2 * TILE_F_)          /* double buffered per wave     */

#define NPART_ (B_ * HKV_ * SPLITS_) /* 1024 partials */
#define WS_M_ 0                      /* [NPART_][4] running max  */
#define WS_S_ (NPART_ * 4)           /* [NPART_][4] denom        */
#define WS_O_ (2 * NPART_ * 4)       /* [NPART_][512] unnorm out */

#define ZERO8 {0.f, 0.f, 0.f, 0.f, 0.f, 0.f, 0.f, 0.f}

/* Broadcast P^T[key, head=lane&15] (WMMA D layout p8[8]) to both lane halves.
   key is a compile-time constant after unrolling. */
#define ROWV(key, dst)                                                     \
  {                                                                        \
    float v_ = p8[(key) & 7];                                              \
    float o_ = __shfl_xor(v_, 16, 32);                                     \
    dst = (((key) < 8) == (q_ == 0)) ? v_ : o_;                            \
  }

/* Issue one page (16 keys x 128 f32) of K and V as 32 async LDS copies.
   Per row: 32 lanes x b128 = 512B, fully coalesced, non-temporal (read-once
   stream, don't displace L2). Tracked by ASYNCcnt, completes in order. */
__device__ __forceinline__ void stage_async(
    const float* __restrict__ kcache, const float* __restrict__ vcache,
    const float* __restrict__ knew, const float* __restrict__ vnew, int b,
    int h, int page, int c, int L, unsigned kloff0, unsigned vloff0,
    unsigned goff) {
  const size_t kb = ((size_t)page * PAGE_) * (HKV_ * D_) + (size_t)h * D_;
  const size_t nb = (size_t)b * (HKV_ * D_) + (size_t)h * D_;
#pragma unroll
  for (int i = 0; i < 16; ++i) {
    const int pos = (c << 4) + i;
    const bool isnew = (pos == L - 1); /* reference scatters new token here */
    const float* ks = isnew ? (knew + nb) : (kcache + kb + (size_t)i * (HKV_ * D_));
    const float* vs = isnew ? (vnew + nb) : (vcache + kb + (size_t)i * (HKV_ * D_));
    const unsigned kl = kloff0 + (unsigned)(i * (KROW_ * 4)) + goff;
    const unsigned vl = vloff0 + (unsigned)(i * (VROW_ * 4)) + goff;
    asm volatile("global_load_async_to_lds_b128 %0, %1, %2 th:TH_LOAD_NT"
                 :
                 : "v"(kl), "v"(goff), "s"(ks)
                 : "memory");
    asm volatile("global_load_async_to_lds_b128 %0, %1, %2 th:TH_LOAD_NT"
                 :
                 : "v"(vl), "v"(goff), "s"(vs)
                 : "memory");
  }
}

__global__ __launch_bounds__(128) void radix_attn_partial(
    const float* __restrict__ q, const float* __restrict__ knew,
    const float* __restrict__ vnew, const float* __restrict__ kcache,
    const float* __restrict__ vcache, const int* __restrict__ page_table,
    const int* __restrict__ seq_lens, float* __restrict__ ws) {
  extern __shared__ float smem[];
  const int bh = blockIdx.x;
  const int b = bh >> 3, h = bh & 7;
  const int split = blockIdx.y;
  const int tid = threadIdx.x;
  const int wave = tid >> 5, lane = tid & 31;
  const int q_ = lane >> 4, m16 = lane & 15;

  const int L = seq_lens[b];
  const int nchunks = (L + 15) >> 4;
  const int c0 = split * CPS_;
  const int c1 = (nchunks < c0 + CPS_) ? nchunks : (c0 + CPS_);

  float* buf0 = smem + wave * WAVE_LDS_;
  float* buf1 = buf0 + TILE_F_;
  /* flat LDS pointer low 32 bits == byte offset from the wave's LDS base
     (aperture bits live in [63:32]); async-to-LDS adds LDS_BASE itself. */
  const unsigned lb0 = (unsigned)(uintptr_t)buf0;
  const unsigned lb1 = (unsigned)(uintptr_t)buf1;
  const unsigned goff = (unsigned)(lane * 16);

  /* ---- Q B-fragments (scores^T = K_tile x Q^T), resident in VGPRs.
     B 4x16 layout: VGPR0 lanes0-15:K=0 / lanes16-31:K=2, VGPR1: K=1 / K=3. */
  v2f bq[32];
  {
    const bool real = (m16 < G_);
    const float* qp =
        q + (size_t)b * (HQ_ * D_) + (size_t)(h * G_ + (m16 & 3)) * D_ + 2 * q_;
#pragma unroll
    for (int c2 = 0; c2 < 32; ++c2) {
      v2f t = *(const v2f*)(qp + 4 * c2);
      v2f z;
      z.x = real ? t.x : 0.0f;
      z.y = real ? t.y : 0.0f;
      bq[c2] = z;
    }
  }

  v8f acc[8];
#pragma unroll
  for (int t = 0; t < 8; ++t) acc[t] = (v8f)ZERO8;
  float m_run = -3.0e38f, s_run = 0.0f;

  const int cfirst = c0 + wave;
  if (cfirst < c1) {
    stage_async(kcache, vcache, knew, vnew, b, h,
                page_table[b * PPS_ + cfirst], cfirst, L, lb0,
                lb0 + 16 * KROW_ * 4, goff);
  }

  int buf = 0;
  for (int c = cfirst; c < c1; c += 4, buf ^= 1) {
    const bool have_next = (c + 4 < c1);
    if (have_next) {
      /* Async LDS writes are UNORDERED vs pending DS reads; make sure the
         previous tile's LDS reads (same buffer we are about to refill) have
         fully completed before issuing the next async batch. */
      asm volatile("s_wait_dscnt 0" ::: "memory");
      const unsigned nb = buf ? lb0 : lb1;
      stage_async(kcache, vcache, knew, vnew, b, h,
                  page_table[b * PPS_ + c + 4], c + 4, L, nb,
                  nb + 16 * KROW_ * 4, goff);
      /* next page pre-touch two tiles ahead */
      if (c + 8 < c1) {
        const int np = page_table[b * PPS_ + c + 8];
        const size_t pb =
            ((size_t)np * PAGE_) * (HKV_ * D_) + (size_t)h * D_ +
            (size_t)(lane >> 1) * (HKV_ * D_) + (size_t)(lane & 1) * 64;
        __builtin_prefetch(kcache + pb, 0, 1);
        __builtin_prefetch(vcache + pb, 0, 1);
      }
      /* async completes in order: <=32 outstanding => current tile landed */
      asm volatile("s_wait_asynccnt 32" ::: "memory");
    } else {
      asm volatile("s_wait_asynccnt 0" ::: "memory");
    }

    const float* Klds = buf ? buf1 : buf0;
    const float* Vl = Klds + 16 * KROW_;

    /* ---- scores^T[16 keys, 16 heads(4 real)] = K_tile(16x128) x Q^T(128x16) */
    v8f sc = (v8f)ZERO8;
    const float* arow = Klds + m16 * KROW_ + 2 * q_;
#pragma unroll
    for (int c2 = 0; c2 < 32; ++c2) {
      v2f a = *(const v2f*)(arow + 4 * c2);
      sc = __builtin_amdgcn_wmma_f32_16x16x4_f32(false, a, false, bq[c2],
                                                 (short)0, sc, false, false);
    }

    /* ---- scale, logit-cap, mask, online softmax */
    float p8[8];
    float mt = -3.0e38f;
#pragma unroll
    for (int r = 0; r < 8; ++r) {
      const int pos = (c << 4) + r + 8 * q_;
      float val = sc[r] * SCALING_;
      val = CAP_ * tanhf(val * (1.0f / CAP_));
      val = (pos < L) ? val : -3.0e38f;
      p8[r] = val;
      mt = fmaxf(mt, val);
    }
    mt = fmaxf(mt, __shfl_xor(mt, 16, 32));
    const float mn = fmaxf(m_run, mt);
    const float corr = __expf(m_run - mn);
    float ts = 0.0f;
#pragma unroll
    for (int r = 0; r < 8; ++r) {
      p8[r] = __expf(p8[r] - mn);
      ts += p8[r];
    }
    ts += __shfl_xor(ts, 16, 32);
    s_run = s_run * corr + ts;
    m_run = mn;
#pragma unroll
    for (int t = 0; t < 8; ++t) acc[t] = acc[t] * corr;

    /* ---- out^T[128, 16 heads] += V^T(128x16) x P^T(16x16), 4-key K-chunks.
       A-frag: lane n -> V[key=4kk+2q+{0,1}][dt*16+m16]; both b32 reads share a
       base -> ds_load_2addr_b32, pairing movs dual-issue with WMMA. */
#pragma unroll
    for (int kk = 0; kk < 4; ++kk) {
      float r0, r1, r2, r3;
      ROWV(4 * kk + 0, r0);
      ROWV(4 * kk + 1, r1);
      ROWV(4 * kk + 2, r2);
      ROWV(4 * kk + 3, r3);
      v2f pb;
      pb.x = q_ ? r2 : r0;
      pb.y = q_ ? r3 : r1;
      const float* vbase = Vl + (4 * kk + 2 * q_) * VROW_ + m16;
#pragma unroll
      for (int dt = 0; dt < 8; ++dt) {
        v2f va;
        va.x = vbase[dt * 16];
        va.y = vbase[dt * 16 + VROW_];
        acc[dt] = __builtin_amdgcn_wmma_f32_16x16x4_f32(false, va, false, pb,
                                                        (short)0, acc[dt],
                                                        false, false);
      }
    }
  }

  /* ---- merge the 4 wave-partials of this split via LDS (aliases wave0 buf0) */
  __syncthreads();
  float* cmM = smem;       /* [4 waves][4 heads] */
  float* cmS = smem + 16;  /* [4 waves][4 heads] */
  float* outw = smem + 32; /* [4 waves][512]     */
  if (lane < 4) {
    cmM[wave * 4 + lane] = m_run;
    cmS[wave * 4 + lane] = s_run;
  }
  if (m16 < G_) {
#pragma unroll
    for (int dt = 0; dt < 8; ++dt)
#pragma unroll
      for (int r = 0; r < 8; ++r)
        outw[wave * 512 + m16 * 128 + dt * 16 + q_ * 8 + r] = acc[dt][r];
  }
  __syncthreads();

  {
    const int f0 = tid * 4; /* 128 threads x 4 floats = 512 outputs */
    const int hh = f0 >> 7; /* head, uniform per thread */
    float M = -3.0e38f;
#pragma unroll
    for (int w = 0; w < 4; ++w) M = fmaxf(M, cmM[w * 4 + hh]);
    float e[4];
    float den = 0.0f;
#pragma unroll
    for (int w = 0; w < 4; ++w) {
      e[w] = __expf(cmM[w * 4 + hh] - M);
      den += cmS[w * 4 + hh] * e[w];
    }
    const size_t pbase = (size_t)(bh * SPLITS_ + split);
    float* o_ws = ws + WS_O_;
#pragma unroll
    for (int j = 0; j < 4; ++j) {
      float o = 0.0f;
#pragma unroll
      for (int w = 0; w < 4; ++w) o += e[w] * outw[w * 512 + f0 + j];
      o_ws[pbase * 512 + f0 + j] = o;
    }
    if ((tid & 31) == 0) {
      ws[WS_M_ + pbase * 4 + hh] = M;
      ws[WS_S_ + pbase * 4 + hh] = den;
    }
  }
}

__global__ __launch_bounds__(128) void radix_attn_combine(
    const float* __restrict__ ws, float* __restrict__ out) {
  const int bh = blockIdx.x;
  const int tid = threadIdx.x;
  const int f0 = tid * 4;
  const int hh = f0 >> 7;
  const float* m_ws = ws + WS_M_;
  const float* s_ws = ws + WS_S_;
  const float* o_ws = ws + WS_O_;
  float M = -3.0e38f;
#pragma unroll
  for (int p = 0; p < SPLITS_; ++p)
    M = fmaxf(M, m_ws[(bh * SPLITS_ + p) * 4 + hh]);
  float e[SPLITS_];
  float den = 0.0f;
#pragma unroll
  for (int p = 0; p < SPLITS_; ++p) {
    e[p] = __expf(m_ws[(bh * SPLITS_ + p) * 4 + hh] - M);
    den += s_ws[(bh * SPLITS_ + p) * 4 + hh] * e[p];
  }
  const float inv = 1.0f / den;
#pragma unroll
  for (int j = 0; j < 4; ++j) {
    float o = 0.0f;
#pragma unroll
    for (int p = 0; p < SPLITS_; ++p)
      o += e[p] * o_ws[(size_t)(bh * SPLITS_ + p) * 512 + f0 + j];
    /* out[b][(h*G+g)*128+d] == out[bh*512 + f0+j] */
    out[(size_t)bh * 512 + f0 + j] = o * inv;
  }
}

extern "C" void kernel_launch(void* const* d_in, const int* in_sizes, int n_in,
                              void* d_out, int out_size, void* d_ws,
                              size_t ws_size, hipStream_t stream) {
  const float* q = (const float*)d_in[0];
  const float* k = (const float*)d_in[1];
  const float* v = (const float*)d_in[2];
  const float* kc = (const float*)d_in[3];
  const float* vc = (const float*)d_in[4];
  const int* pt = (const int*)d_in[5];
  const int* sl = (const int*)d_in[6];
  float* ws = (float*)d_ws;
  (void)in_sizes; (void)n_in; (void)out_size; (void)ws_size;

  const size_t shmem = (size_t)4 * WAVE_LDS_ * sizeof(float); /* 137216 B (<320KB/WGP) */
  radix_attn_partial<<<dim3(B_ * HKV_, SPLITS_), 128, shmem, stream>>>(
      q, k, v, kc, vc, pt, sl, ws);
  radix_attn_combine<<<dim3(B_ * HKV_), 128, 0, stream>>>(ws, (float*)d_out);
}